// RollingStatsLayer_6554120093995
// MI455X (gfx1250) — compile-verified
//
#include <hip/hip_runtime.h>
#include <cstdint>
#include <cstddef>

#ifndef __has_builtin
#define __has_builtin(x) 0
#endif

#define T_IN   300000
#define NF     64
#define WND    64
#define T_OUT  (T_IN - WND + 1)   // 299937
#define NCOL   (6 * NF)           // 384
#define LDS_ROWS 128

typedef __attribute__((ext_vector_type(2))) float v2f;
typedef __attribute__((ext_vector_type(8))) float v8f;

#if __has_builtin(__builtin_amdgcn_wmma_f32_16x16x4_f32)
#define USE_WMMA 1
#else
#define USE_WMMA 0
#endif

#if __has_builtin(__builtin_amdgcn_global_load_async_to_lds_b128) && \
    __has_builtin(__builtin_amdgcn_s_wait_asynccnt)
#define USE_ASYNC 1
// Exact parameter types per the diagnostic: pointer to 16-byte int vector,
// global (as1) source and LDS (as3) destination.
typedef int v4i_vs __attribute__((vector_size(16)));
typedef __attribute__((address_space(1))) v4i_vs* as1_v4i;
typedef __attribute__((address_space(3))) v4i_vs* as3_v4i;
#else
#define USE_ASYNC 0
#endif

__global__ __launch_bounds__(256)
void rolling_stats_kernel(const float* __restrict__ x, float* __restrict__ out) {
    __shared__ float xs[LDS_ROWS * NF];   // 32 KB input tile: rows t0..t0+127
    __shared__ float sMin[64 * NF];       // 16 KB suffix-min scan
    __shared__ float sMax[64 * NF];       // 16 KB suffix-max scan

    const int tid = threadIdx.x;
    const int t0  = blockIdx.x * 64;      // first output row of this tile

    // ---------------- Phase 0: global -> LDS (128 rows x 64 feats) ----------
    // 2048 float4 transfers, 8 per thread.
    for (int it = 0; it < 8; ++it) {
        int idx = tid + it * 256;         // 0..2047
        int r   = idx >> 4;               // row in tile 0..127
        int c4  = idx & 15;               // float4 column
        int tr  = t0 + r;
        if (tr > T_IN - 1) tr = T_IN - 1; // clamp tail reads
        const float* gp = x + (size_t)tr * NF + c4 * 4;
        float*       lp = &xs[r * NF + c4 * 4];
#if USE_ASYNC
        __builtin_amdgcn_global_load_async_to_lds_b128(
            (as1_v4i)(gp), (as3_v4i)(lp), 0, 0);
#else
        *(float4*)lp = *(const float4*)gp;
#endif
    }
#if USE_ASYNC
    __builtin_amdgcn_s_wait_asynccnt(0);
#endif
    __syncthreads();

    // ---------------- Phase 2a: suffix scans for min/max ---------------------
    // Window [r, r+63] = suffix(block0 rows r..63) U prefix(block1 rows 0..r-1).
    // 128 threads: tid&63 = feature, tid>>6 selects min vs max.
    if (tid < 128) {
        const int  f  = tid & 63;
        const bool mx = (tid >= 64);
        float* scan = mx ? sMax : sMin;
        float run = xs[63 * NF + f];
        scan[63 * NF + f] = run;
        for (int r = 62; r >= 0; --r) {
            float v = xs[r * NF + f];
            run = mx ? fmaxf(run, v) : fminf(run, v);
            scan[r * NF + f] = run;
        }
    }

    // ---------------- Phase 1: rolling sum / sumsq via f32 WMMA --------------
    const int lane = tid & 31;
    const int wave = tid >> 5;
    const int hi   = lane >> 4;   // lane group (0: lanes 0-15, 1: lanes 16-31)
    const int lo   = lane & 15;   // M (A) / N (B,C,D) index

#if USE_WMMA
    // 16 output tiles (4 row-chunks x 4 feature-chunks), 2 per wave.
    for (int tt = wave; tt < 16; tt += 8) {
        const int i0 = (tt & 3) * 16;   // output row chunk within tile
        const int f0 = (tt >> 2) * 16;  // feature chunk
        v8f acc1 = {};                  // rolling sum
        v8f acc2 = {};                  // rolling sum of squares
        #pragma unroll
        for (int j = 0; j < 20; ++j) {
            // A(16x4) layout: element(v, lane) -> M = lo, K = v + 2*hi
            // banded ones: A=1 iff 0 <= (4j + K) - M <= 63
            const int k0 = 4 * j + 2 * hi;
            const int u0 = k0 - lo;
            const int u1 = k0 + 1 - lo;
            v2f a;
            a.x = (u0 >= 0 && u0 < WND) ? 1.0f : 0.0f;
            a.y = (u1 >= 0 && u1 < WND) ? 1.0f : 0.0f;
            // B(4x16) layout: element(v, lane) -> K = v + 2*hi, N = lo
            v2f b;
            b.x = xs[(i0 + k0)     * NF + f0 + lo];
            b.y = xs[(i0 + k0 + 1) * NF + f0 + lo];
            acc1 = __builtin_amdgcn_wmma_f32_16x16x4_f32(
                false, a, false, b, (short)0, acc1, false, false);
            v2f b2 = b * b;
            acc2 = __builtin_amdgcn_wmma_f32_16x16x4_f32(
                false, a, false, b2, (short)0, acc2, false, false);
        }
        // C/D layout: VGPR r, lane -> M = r + 8*hi, N = lo
        #pragma unroll
        for (int r = 0; r < 8; ++r) {
            int i  = i0 + r + 8 * hi;   // output row within tile
            int to = t0 + i;
            if (to < T_OUT) {
                int   f    = f0 + lo;
                float s1   = acc1[r];
                float s2   = acc2[r];
                float mean = s1 * (1.0f / 64.0f);
                float var  = fmaxf(s2 * (1.0f / 64.0f) - mean * mean, 0.0f);
                float sd   = sqrtf(var);
                float orig = xs[(i + 63) * NF + f];
                float* o = out + (size_t)to * NCOL;
                o[f]        = orig;   // original (trimmed)
                o[64  + f]  = mean;   // mean
                o[128 + f]  = sd;     // std (population)
                o[320 + f]  = s1;     // rolling sum
            }
        }
    }
#else
    // Scalar fallback (keeps the kernel compiling if the builtin is absent).
    for (int p = tid; p < 64 * NF; p += 256) {
        int i = p >> 6, f = p & 63;
        int to = t0 + i;
        float s1 = 0.0f, s2 = 0.0f;
        for (int w = 0; w < WND; ++w) {
            float v = xs[(i + w) * NF + f];
            s1 += v; s2 += v * v;
        }
        if (to < T_OUT) {
            float mean = s1 * (1.0f / 64.0f);
            float var  = fmaxf(s2 * (1.0f / 64.0f) - mean * mean, 0.0f);
            float* o = out + (size_t)to * NCOL;
            o[f]       = xs[(i + 63) * NF + f];
            o[64 + f]  = mean;
            o[128 + f] = sqrtf(var);
            o[320 + f] = s1;
        }
    }
#endif

    __syncthreads();

    // ---------------- Phase 2b: prefix + combine + store min/max -------------
    if (tid < 128) {
        const int   f    = tid & 63;
        const bool  mx   = (tid >= 64);
        const float* scan = mx ? sMax : sMin;
        const int   col  = (mx ? 256 : 192) + f;
        if (t0 < T_OUT)
            out[(size_t)t0 * NCOL + col] = scan[f];   // r=0: whole first block
        float p = 0.0f;
        for (int r = 1; r < 64; ++r) {
            float xv = xs[(63 + r) * NF + f];         // prefix element of block1
            p = (r == 1) ? xv : (mx ? fmaxf(p, xv) : fminf(p, xv));
            float s   = scan[r * NF + f];
            float res = mx ? fmaxf(s, p) : fminf(s, p);
            int to = t0 + r;
            if (to < T_OUT)
                out[(size_t)to * NCOL + col] = res;
        }
    }
}

extern "C" void kernel_launch(void* const* d_in, const int* in_sizes, int n_in,
                              void* d_out, int out_size, void* d_ws, size_t ws_size,
                              hipStream_t stream) {
    (void)in_sizes; (void)n_in; (void)d_ws; (void)ws_size; (void)out_size;
    const float* x = (const float*)d_in[0];
    float* out = (float*)d_out;
    const int nblocks = (T_OUT + 63) / 64;  // 4687
    rolling_stats_kernel<<<dim3(nblocks), dim3(256), 0, stream>>>(x, out);
}